// LinearShift_46866683134638
// MI455X (gfx1250) — compile-verified
//
#include <hip/hip_runtime.h>

typedef __attribute__((ext_vector_type(2))) float v2f;
typedef __attribute__((ext_vector_type(4))) float v4f;
typedef __attribute__((ext_vector_type(8))) float v8f;

#define IN_F 128
#define OUT_F 128
#define LDS_STRIDE 132           // 128 + 4 pad -> conflict-free strided b64 reads
#define ROWS_PER_BLOCK 128
#define BLOCK_THREADS 256

// deepshift round_to_fixed: floor to 2^-16 grid, clamp to [-2^15, 2^15 - 1]
__device__ __forceinline__ float quant_fixed(float x) {
    float r = floorf(x * 65536.0f) * (1.0f / 65536.0f);
    return fminf(fmaxf(r, -32768.0f), 32767.0f);
}

__global__ __launch_bounds__(BLOCK_THREADS)
void linear_shift_wmma(const float* __restrict__ x,
                       const float* __restrict__ shift,
                       const float* __restrict__ sgn,
                       const float* __restrict__ bias,
                       float* __restrict__ out,
                       int nrows) {
    extern __shared__ float smem[];
    float* Ash = smem;                                   // [128][132] quantized x tile
    float* Bsh = smem + ROWS_PER_BLOCK * LDS_STRIDE;     // [128][132] quantized weights v[out][in]

    const int tid  = threadIdx.x;
    const int lane = tid & 31;
    const int wave = tid >> 5;
    const int mrow = lane & 15;        // M row (0..15) within wave tile / N col within 16-tile
    const int hi   = lane >> 4;        // 0: K=0,1 half  1: K=2,3 half
    const long rowBase = (long)blockIdx.x * ROWS_PER_BLOCK;

    // ---- stage quantized shift-weights v[out][in] into LDS (coalesced) ----
    for (int i = 0; i < (OUT_F * IN_F) / BLOCK_THREADS; ++i) {
        int e = tid + i * BLOCK_THREADS;          // e = out*128 + in
        int o = e >> 7;
        int k = e & 127;
        float sr = rintf(fminf(fmaxf(shift[e], -14.0f), 0.0f));
        float sg = rintf(sgn[e]);
        sg = (sg > 0.0f) ? 1.0f : ((sg < 0.0f) ? -1.0f : 0.0f);
        Bsh[o * LDS_STRIDE + k] = exp2f(sr) * sg;
    }

    // ---- stage quantized A tile into LDS (float4 coalesced global loads) ----
    const float* xblk = x + rowBase * IN_F;
    for (int i = 0; i < (ROWS_PER_BLOCK * IN_F) / (BLOCK_THREADS * 4); ++i) {
        int e4 = (tid + i * BLOCK_THREADS) * 4;   // flat float index in 128x128 tile
        int r  = e4 >> 7;
        int c  = e4 & 127;
        v4f v = *(const v4f*)(xblk + e4);
        float* dst = Ash + r * LDS_STRIDE + c;
        dst[0] = quant_fixed(v[0]);
        dst[1] = quant_fixed(v[1]);
        dst[2] = quant_fixed(v[2]);
        dst[3] = quant_fixed(v[3]);
    }
    __syncthreads();

    // ---- accumulators: 16 rows x 128 cols per wave, bias-initialized ----
    v8f acc[8];
#pragma unroll
    for (int n = 0; n < 8; ++n) {
        float bq = quant_fixed(bias[n * 16 + mrow]);
#pragma unroll
        for (int j = 0; j < 8; ++j) acc[n][j] = bq;
    }

    // A fragment base: lane (hi,mrow) holds A[M=mrow][K = 4k + 2*hi + {0,1}]
    const float* aptr = Ash + (wave * 16 + mrow) * LDS_STRIDE + 2 * hi;
    // B fragment base: VGPR j holds B[K = 4k + 2*hi + j][N = mrow] = v[N][K]
    const float* bptr = Bsh + mrow * LDS_STRIDE + 2 * hi;

    for (int k = 0; k < IN_F / 4; ++k) {
        v2f a = *(const v2f*)(aptr + 4 * k);
#pragma unroll
        for (int n = 0; n < 8; ++n) {
            v2f b = *(const v2f*)(bptr + n * 16 * LDS_STRIDE + 4 * k);
            // D = A(16x4,f32) * B(4x16,f32) + C(16x16,f32)
            acc[n] = __builtin_amdgcn_wmma_f32_16x16x4_f32(
                /*neg_a=*/false, a, /*neg_b=*/false, b,
                /*c_mod=*/(short)0, acc[n], /*reuse_a=*/false, /*reuse_b=*/false);
        }
    }

    // ---- store: VGPR j -> row (j + 8*hi), lane mrow -> col (n*16 + mrow) ----
    float* orow = out + (rowBase + wave * 16 + hi * 8) * OUT_F + mrow;
#pragma unroll
    for (int n = 0; n < 8; ++n) {
#pragma unroll
        for (int j = 0; j < 8; ++j) {
            orow[(long)j * OUT_F + n * 16] = acc[n][j];
        }
    }
    (void)nrows;
}

extern "C" void kernel_launch(void* const* d_in, const int* in_sizes, int n_in,
                              void* d_out, int out_size, void* d_ws, size_t ws_size,
                              hipStream_t stream) {
    const float* x     = (const float*)d_in[0];
    const float* shift = (const float*)d_in[1];
    const float* sgn   = (const float*)d_in[2];
    const float* bias  = (const float*)d_in[3];
    float* out = (float*)d_out;

    int nrows = in_sizes[0] / IN_F;                 // 524288
    int grid  = nrows / ROWS_PER_BLOCK;             // 4096
    size_t lds_bytes = 2u * ROWS_PER_BLOCK * LDS_STRIDE * sizeof(float);  // 132 KB

    hipFuncSetAttribute(reinterpret_cast<const void*>(linear_shift_wmma),
                        hipFuncAttributeMaxDynamicSharedMemorySize,
                        (int)lds_bytes);

    linear_shift_wmma<<<grid, BLOCK_THREADS, lds_bytes, stream>>>(
        x, shift, sgn, bias, out, nrows);
}